// MolecularGNN_39556648796267
// MI455X (gfx1250) — compile-verified
//
#include <hip/hip_runtime.h>
#include <hip/hip_bf16.h>

typedef __attribute__((ext_vector_type(2))) float v2f;
typedef __attribute__((ext_vector_type(8))) float v8f;

enum : int {
    N_NODES = 50000,
    N_EDGES = 600000,
    N_GRAPHS = 64,
    D_NODE = 64,
    HID = 128,
    N_LAYERS = 3
};
#define LN_EPS 1e-5f

// ---------------------------------------------------------------------------
// Utility: zero-fill a float buffer
// ---------------------------------------------------------------------------
__global__ void zero_f(float* __restrict__ p, int n) {
    int i = blockIdx.x * blockDim.x + threadIdx.x;
    if (i < n) p[i] = 0.0f;
}

// ---------------------------------------------------------------------------
// Degree: deg[dst] += 1 over edges, then dis = rsqrt(deg + 1)
// ---------------------------------------------------------------------------
__global__ void deg_count(const int* __restrict__ dst, float* __restrict__ deg) {
    int e = blockIdx.x * blockDim.x + threadIdx.x;
    if (e < N_EDGES) atomicAdd(&deg[dst[e]], 1.0f);
}

__global__ void deg_finish(float* __restrict__ dis) {
    int i = blockIdx.x * blockDim.x + threadIdx.x;
    if (i < N_NODES) dis[i] = rsqrtf(dis[i] + 1.0f);
}

// ---------------------------------------------------------------------------
// Node encoder: h = LayerNorm(ReLU(x @ enc_w + enc_b))
// One wave (32 threads) per 16-row tile. WMMA f32 16x16x4.
//   A lane layout: lane L -> row L%16, K pair = 2*(L/16)
//   C/D lane layout: lane L, vgpr v -> row v + 8*(L/16), col L%16
// ---------------------------------------------------------------------------
__global__ __launch_bounds__(32) void encoder_kernel(
    const float* __restrict__ x, const float* __restrict__ W,
    const float* __restrict__ bias, const float* __restrict__ g,
    const float* __restrict__ bln, float* __restrict__ h)
{
    const int row0 = blockIdx.x * 16;
    const int lane = threadIdx.x;
    const int m = lane & 15;
    const int khalf = lane >> 4;

    v8f acc[8];
    for (int nt = 0; nt < 8; ++nt) acc[nt] = (v8f)0.0f;

    const int arow = row0 + m;
    for (int k = 0; k < D_NODE; k += 4) {
        const int ka = k + khalf * 2;
        v2f a;
        a.x = x[arow * D_NODE + ka];
        a.y = x[arow * D_NODE + ka + 1];
        for (int nt = 0; nt < 8; ++nt) {
            const int n = nt * 16 + m;
            v2f b;
            b.x = W[(ka + 0) * HID + n];
            b.y = W[(ka + 1) * HID + n];
            acc[nt] = __builtin_amdgcn_wmma_f32_16x16x4_f32(
                false, a, false, b, (short)0, acc[nt], false, false);
        }
    }

    // bias + ReLU, accumulate LayerNorm stats per accumulator row slot
    float s[8], q[8];
    for (int v = 0; v < 8; ++v) { s[v] = 0.0f; q[v] = 0.0f; }
    for (int nt = 0; nt < 8; ++nt) {
        const float bb = bias[nt * 16 + m];
        for (int v = 0; v < 8; ++v) {
            float t = acc[nt][v] + bb;
            t = fmaxf(t, 0.0f);
            acc[nt][v] = t;
            s[v] += t;
            q[v] += t * t;
        }
    }
    // reduce across the 16 lanes of each half (row features live in one half)
    for (int mask = 1; mask <= 8; mask <<= 1) {
        for (int v = 0; v < 8; ++v) {
            s[v] += __shfl_xor(s[v], mask, 32);
            q[v] += __shfl_xor(q[v], mask, 32);
        }
    }
    float mu[8], rstd[8];
    for (int v = 0; v < 8; ++v) {
        mu[v] = s[v] * (1.0f / HID);
        float var = q[v] * (1.0f / HID) - mu[v] * mu[v];
        rstd[v] = rsqrtf(var + LN_EPS);
    }
    const int rhalf = lane >> 4;
    for (int nt = 0; nt < 8; ++nt) {
        const int n = nt * 16 + m;
        const float gg = g[n], bb2 = bln[n];
        for (int v = 0; v < 8; ++v) {
            const int row = row0 + v + 8 * rhalf;
            h[row * HID + n] = (acc[nt][v] - mu[v]) * rstd[v] * gg + bb2;
        }
    }
}

// ---------------------------------------------------------------------------
// Conv GEMM: hw = h @ W   (M=50000, K=128, N=128), WMMA f32 16x16x4
// ---------------------------------------------------------------------------
__global__ __launch_bounds__(32) void conv_mm(
    const float* __restrict__ h, const float* __restrict__ W,
    float* __restrict__ hw)
{
    const int row0 = blockIdx.x * 16;
    const int lane = threadIdx.x;
    const int m = lane & 15;
    const int khalf = lane >> 4;

    v8f acc[8];
    for (int nt = 0; nt < 8; ++nt) acc[nt] = (v8f)0.0f;

    const int arow = row0 + m;
    for (int k = 0; k < HID; k += 4) {
        const int ka = k + khalf * 2;
        v2f a;
        a.x = h[arow * HID + ka];
        a.y = h[arow * HID + ka + 1];
        for (int nt = 0; nt < 8; ++nt) {
            const int n = nt * 16 + m;
            v2f b;
            b.x = W[(ka + 0) * HID + n];
            b.y = W[(ka + 1) * HID + n];
            acc[nt] = __builtin_amdgcn_wmma_f32_16x16x4_f32(
                false, a, false, b, (short)0, acc[nt], false, false);
        }
    }
    const int rhalf = lane >> 4;
    for (int nt = 0; nt < 8; ++nt) {
        const int n = nt * 16 + m;
        for (int v = 0; v < 8; ++v) {
            const int row = row0 + v + 8 * rhalf;
            hw[row * HID + n] = acc[nt][v];
        }
    }
}

// ---------------------------------------------------------------------------
// Edge scatter: agg[dst] += hw[src] * dis[src]*dis[dst]
// One 32-lane wave per edge; each lane handles 4 features (float4 load).
// agg/hw are L2-resident (25.6 MB each vs 192 MB L2).
// ---------------------------------------------------------------------------
__global__ void scatter_kernel(const float* __restrict__ hw,
                               const int* __restrict__ src,
                               const int* __restrict__ dst,
                               const float* __restrict__ dis,
                               float* __restrict__ agg)
{
    const int e = blockIdx.x * (blockDim.x >> 5) + (threadIdx.x >> 5);
    if (e >= N_EDGES) return;
    const int lane = threadIdx.x & 31;
    const int s = src[e];
    const int d = dst[e];
    const float w = dis[s] * dis[d];
    const float4 v = ((const float4*)(hw + s * HID))[lane];
    float* base = agg + d * HID + lane * 4;
    atomicAdd(base + 0, v.x * w);
    atomicAdd(base + 1, v.y * w);
    atomicAdd(base + 2, v.z * w);
    atomicAdd(base + 3, v.w * w);
}

// ---------------------------------------------------------------------------
// Combine: h = relu(agg + hw * dis^2 + bias)
// ---------------------------------------------------------------------------
__global__ void combine_kernel(float* __restrict__ h,
                               const float* __restrict__ agg,
                               const float* __restrict__ hw,
                               const float* __restrict__ dis,
                               const float* __restrict__ bias)
{
    const int i = blockIdx.x * blockDim.x + threadIdx.x;
    if (i >= N_NODES * HID) return;
    const int node = i >> 7;
    const int f = i & (HID - 1);
    const float sn = dis[node] * dis[node];
    const float v = agg[i] + hw[i] * sn + bias[f];
    h[i] = fmaxf(v, 0.0f);
}

// ---------------------------------------------------------------------------
// Pooling: per-graph sums and counts
// ---------------------------------------------------------------------------
__global__ void pool_cnt(const int* __restrict__ batch, float* __restrict__ cnt) {
    const int n = blockIdx.x * blockDim.x + threadIdx.x;
    if (n < N_NODES) atomicAdd(&cnt[batch[n]], 1.0f);
}

__global__ void pool_sum(const float* __restrict__ h,
                         const int* __restrict__ batch,
                         float* __restrict__ pooled)
{
    const int i = blockIdx.x * blockDim.x + threadIdx.x;
    if (i >= N_NODES * HID) return;
    const int node = i >> 7;
    const int f = i & (HID - 1);
    atomicAdd(&pooled[batch[node] * HID + f], h[i]);
}

// ---------------------------------------------------------------------------
// Classifier
// ---------------------------------------------------------------------------
__global__ void cls1(const float* __restrict__ pooled,
                     const float* __restrict__ cnt,
                     const float* __restrict__ w,   // [128,64]
                     const float* __restrict__ b,   // [64]
                     float* __restrict__ t1)        // [64,64]
{
    const int idx = blockIdx.x * blockDim.x + threadIdx.x;
    if (idx >= N_GRAPHS * 64) return;
    const int gph = idx >> 6;
    const int j = idx & 63;
    float acc = 0.0f;
    for (int f = 0; f < HID; ++f)
        acc += pooled[gph * HID + f] * w[f * 64 + j];
    acc /= fmaxf(cnt[gph], 1.0f);
    t1[idx] = fmaxf(acc + b[j], 0.0f);
}

__global__ void cls2(const float* __restrict__ t1,
                     const float* __restrict__ w,   // [64,1]
                     const float* __restrict__ b,   // [1]
                     float* __restrict__ out)       // [64]
{
    const int gph = threadIdx.x;
    if (gph >= N_GRAPHS) return;
    float acc = 0.0f;
    for (int j = 0; j < 64; ++j)
        acc += t1[gph * 64 + j] * w[j];
    out[gph] = acc + b[0];
}

// ---------------------------------------------------------------------------
// Launcher
// ---------------------------------------------------------------------------
extern "C" void kernel_launch(void* const* d_in, const int* in_sizes, int n_in,
                              void* d_out, int out_size, void* d_ws, size_t ws_size,
                              hipStream_t stream) {
    (void)in_sizes; (void)n_in; (void)out_size; (void)ws_size;

    const float* x      = (const float*)d_in[0];
    const int*   eidx   = (const int*)d_in[1];
    const int*   src    = eidx;
    const int*   dst    = eidx + N_EDGES;
    const int*   batch  = (const int*)d_in[3];
    const float* enc_w  = (const float*)d_in[4];
    const float* enc_b  = (const float*)d_in[5];
    const float* ln_g   = (const float*)d_in[6];
    const float* ln_b   = (const float*)d_in[7];
    const float* conv_ws_p = (const float*)d_in[10];  // [3,128,128]
    const float* conv_bs_p = (const float*)d_in[11];  // [3,128]
    const float* c1_w   = (const float*)d_in[12];
    const float* c1_b   = (const float*)d_in[13];
    const float* c2_w   = (const float*)d_in[14];
    const float* c2_b   = (const float*)d_in[15];
    float* out = (float*)d_out;

    // Workspace layout (floats)
    float* ws  = (float*)d_ws;
    float* dis = ws;                       // N_NODES (deg, then rsqrt in place)
    float* h   = dis + N_NODES;            // N_NODES*HID
    float* hw  = h + (size_t)N_NODES * HID;
    float* agg = hw + (size_t)N_NODES * HID;
    float* pooled = agg + (size_t)N_NODES * HID;  // 64*128
    float* cnt = pooled + N_GRAPHS * HID;         // 64
    float* t1  = cnt + N_GRAPHS;                  // 64*64

    const int NH = N_NODES * HID;

    // 1) degree / normalization
    zero_f<<<(N_NODES + 255) / 256, 256, 0, stream>>>(dis, N_NODES);
    deg_count<<<(N_EDGES + 255) / 256, 256, 0, stream>>>(dst, dis);
    deg_finish<<<(N_NODES + 255) / 256, 256, 0, stream>>>(dis);

    // 2) encoder (WMMA + fused ReLU/LayerNorm)
    encoder_kernel<<<N_NODES / 16, 32, 0, stream>>>(x, enc_w, enc_b, ln_g, ln_b, h);

    // 3) GCN layers
    for (int l = 0; l < N_LAYERS; ++l) {
        conv_mm<<<N_NODES / 16, 32, 0, stream>>>(h, conv_ws_p + (size_t)l * HID * HID, hw);
        zero_f<<<(NH + 255) / 256, 256, 0, stream>>>(agg, NH);
        scatter_kernel<<<(N_EDGES + 7) / 8, 256, 0, stream>>>(hw, src, dst, dis, agg);
        combine_kernel<<<(NH + 255) / 256, 256, 0, stream>>>(h, agg, hw, dis,
                                                             conv_bs_p + (size_t)l * HID);
    }

    // 4) global mean pool (pooled + cnt are contiguous -> single zero fill)
    zero_f<<<(N_GRAPHS * HID + N_GRAPHS + 255) / 256, 256, 0, stream>>>(pooled,
                                                                        N_GRAPHS * HID + N_GRAPHS);
    pool_cnt<<<(N_NODES + 255) / 256, 256, 0, stream>>>(batch, cnt);
    pool_sum<<<(NH + 255) / 256, 256, 0, stream>>>(h, batch, pooled);

    // 5) classifier
    cls1<<<(N_GRAPHS * 64 + 255) / 256, 256, 0, stream>>>(pooled, cnt, c1_w, c1_b, t1);
    cls2<<<1, 64, 0, stream>>>(t1, c2_w, c2_b, out);
}